// NMSPostProcess_1975684956495
// MI455X (gfx1250) — compile-verified
//
#include <hip/hip_runtime.h>
#include <stdint.h>

#define BS 4
#define NQ 900
#define NCLS 1203
#define NC (NQ * NCLS)        /* 1,082,700 */
#define K_TOP 10000
#define CAP 16384
#define NSORT 16384
#define IOU_THRESH 0.7f

/* ---- workspace layout (bytes) ---- */
#define WS_HIST   0x000000u   /* u32[BS][65536]          1 MB   */
#define WS_THR    0x100000u   /* u32[BS]                        */
#define WS_SELCNT 0x100040u   /* u32[BS]                        */
#define WS_CANDL  0x100100u   /* float[BS][CAP]          256 KB */
#define WS_CANDI  0x140100u   /* u32[BS][CAP]            256 KB */
#define WS_SBOX   0x180100u   /* float[BS][K_TOP][4]     625 KB */
#define WS_SSCORE 0x21C500u   /* float[BS][K_TOP]        156 KB */
#define WS_SLABEL 0x243600u   /* i32[BS][K_TOP]          156 KB */
#define WS_KEEP   0x26A700u   /* u8[BS][K_TOP]            40 KB */

/* sort kernel dynamic LDS: kv 128KB | candL stage 64KB | candI stage 64KB */
#define SORT_LDS  262144u

/* NMS kernel dynamic LDS layout */
#define L_BOX   0u        /* float4[K_TOP]  160000 B */
#define L_LAB   160000u   /* i32[K_TOP]      40000 B */
#define L_AREA  200000u   /* float[K_TOP]    40000 B */
#define L_KEEP  240000u   /* u8[K_TOP]+pad   10016 B */
#define L_RED   250016u   /* float[1024]      4096 B */
#define NMS_LDS 254112u

__device__ __forceinline__ uint32_t fkey(float f) {
  uint32_t u = __float_as_uint(f);
  return (u & 0x80000000u) ? ~u : (u | 0x80000000u);
}
__device__ __forceinline__ float unkey(uint32_t k) {
  uint32_t u = (k & 0x80000000u) ? (k ^ 0x80000000u) : ~k;
  return __uint_as_float(u);
}
/* CDNA5 async global->LDS helpers (per-lane; tracked by ASYNCcnt) */
__device__ __forceinline__ void async_b128(uint32_t lds_off, const void* g) {
  unsigned long long ga = (unsigned long long)(uintptr_t)g;
  asm volatile("global_load_async_to_lds_b128 %0, %1, off" :: "v"(lds_off), "v"(ga) : "memory");
}
__device__ __forceinline__ void async_b32(uint32_t lds_off, const void* g) {
  unsigned long long ga = (unsigned long long)(uintptr_t)g;
  asm volatile("global_load_async_to_lds_b32 %0, %1, off" :: "v"(lds_off), "v"(ga) : "memory");
}
__device__ __forceinline__ void wait_async0() {
  asm volatile("s_wait_asynccnt 0" ::: "memory");
}

/* ---------------- K0: zero histogram + counters ---------------- */
__global__ void k_zero(uint32_t* __restrict__ w, int n) {
  int stride = gridDim.x * blockDim.x;
  for (int i = blockIdx.x * blockDim.x + threadIdx.x; i < n; i += stride) w[i] = 0u;
}

/* ---------------- K1: 65536-bin histogram of order-keys (batch = blockIdx.y) ------- */
__global__ void k_hist(const float* __restrict__ logits, uint32_t* __restrict__ hist) {
  int b = blockIdx.y;
  const float* src = logits + (size_t)b * NC;
  uint32_t* h = hist + ((uint32_t)b << 16);
  int stride = gridDim.x * blockDim.x;
  for (int i = blockIdx.x * blockDim.x + threadIdx.x; i < NC; i += stride) {
    __builtin_prefetch(&src[i + stride], 0, 0);   /* global_prefetch_b8 */
    atomicAdd(&h[fkey(src[i]) >> 16], 1u);
  }
}

/* ---------------- K2: per-batch threshold bin ---------------- */
__global__ void __launch_bounds__(1024) k_thresh(const uint32_t* __restrict__ hist,
                                                 uint32_t* __restrict__ thr) {
  __shared__ uint32_t sh[1024];
  int b = blockIdx.x, tid = threadIdx.x;
  const uint32_t* h = hist + ((uint32_t)b << 16);
  uint32_t s = 0;
  int base = tid * 64;
  for (int k = 0; k < 64; ++k) s += h[base + k];
  sh[tid] = s;
  __syncthreads();
  if (tid == 0) {
    uint32_t acc = 0;
    uint32_t T = 0;
    for (int c = 1023; c >= 0; --c) {
      if (acc + sh[c] >= (uint32_t)K_TOP || c == 0) {
        uint32_t a2 = acc;
        T = (uint32_t)(c * 64);
        for (int bin = c * 64 + 63; bin >= c * 64; --bin) {
          a2 += h[bin];
          if (a2 >= (uint32_t)K_TOP) { T = (uint32_t)bin; break; }
        }
        break;
      }
      acc += sh[c];
    }
    thr[b] = T;
  }
}

/* ---------------- K3: compact all candidates with bin >= threshold ---------------- */
__global__ void k_compact(const float* __restrict__ logits, const uint32_t* __restrict__ thr,
                          uint32_t* __restrict__ selcnt, float* __restrict__ candL,
                          uint32_t* __restrict__ candI) {
  int b = blockIdx.y;
  const float* src = logits + (size_t)b * NC;
  uint32_t t = thr[b];
  int stride = gridDim.x * blockDim.x;
  for (int i = blockIdx.x * blockDim.x + threadIdx.x; i < NC; i += stride) {
    float x = src[i];
    if ((fkey(x) >> 16) >= t) {
      uint32_t pos = atomicAdd(&selcnt[b], 1u);
      if (pos < (uint32_t)CAP) {
        candL[b * CAP + pos] = x;
        candI[b * CAP + pos] = (uint32_t)i;
      }
    }
  }
}

/* ---------------- K4: in-LDS bitonic sort (exact top-K) + decode ---------------- */
__global__ void __launch_bounds__(1024) k_sort(const float* __restrict__ candL,
                                               const uint32_t* __restrict__ candI,
                                               const uint32_t* __restrict__ selcnt,
                                               const float* __restrict__ pred_boxes,
                                               const float* __restrict__ tsz,
                                               float* __restrict__ sbox,
                                               float* __restrict__ sscore,
                                               int* __restrict__ slabel) {
  extern __shared__ char sms[];
  unsigned long long* kv = (unsigned long long*)sms;          /* NSORT x u64 = 128 KB */
  float* sL = (float*)(sms + 131072);                         /* CAP floats = 64 KB   */
  uint32_t* sI = (uint32_t*)(sms + 131072 + 65536);           /* CAP u32   = 64 KB   */
  int b = blockIdx.x, tid = threadIdx.x;
  uint32_t n = selcnt[b];
  if (n > (uint32_t)CAP) n = CAP;

  /* async-stage both candidate arrays into LDS */
  const float* gL = candL + (size_t)b * CAP;
  const uint32_t* gI = candI + (size_t)b * CAP;
  for (int t = tid * 4; t < CAP; t += 1024 * 4) {
    async_b128((uint32_t)(uintptr_t)(&sL[t]), gL + t);
    async_b128((uint32_t)(uintptr_t)(&sI[t]), gI + t);
  }
  wait_async0();
  __syncthreads();

  for (int t = tid; t < NSORT; t += 1024) {
    unsigned long long v = 0ull;
    if (t < (int)n)
      v = ((unsigned long long)fkey(sL[t]) << 32) | (unsigned long long)sI[t];
    kv[t] = v;
  }
  __syncthreads();
  /* bitonic, final order descending by (key,idx) */
  for (unsigned k = 2; k <= (unsigned)NSORT; k <<= 1) {
    for (unsigned j = k >> 1; j > 0; j >>= 1) {
      for (unsigned i = (unsigned)tid; i < (unsigned)NSORT; i += 1024u) {
        unsigned p = i ^ j;
        if (p > i) {
          unsigned long long a = kv[i], c = kv[p];
          bool up = ((i & k) == 0u);
          if (up ? (a < c) : (a > c)) { kv[i] = c; kv[p] = a; }
        }
      }
      __syncthreads();
    }
  }
  float ih = tsz[b * 2 + 0], iw = tsz[b * 2 + 1];
  for (int r = tid; r < K_TOP; r += 1024) {
    unsigned long long v = kv[r];
    uint32_t idx = (uint32_t)v;
    float logit = unkey((uint32_t)(v >> 32));
    int q = (int)(idx / (uint32_t)NCLS);
    int lab = (int)(idx - (uint32_t)q * (uint32_t)NCLS);
    const float4 bx = *(const float4*)(pred_boxes + ((size_t)b * NQ + (size_t)q) * 4);
    float x1 = (bx.x - 0.5f * bx.z) * iw;
    float y1 = (bx.y - 0.5f * bx.w) * ih;
    float x2 = (bx.x + 0.5f * bx.z) * iw;
    float y2 = (bx.y + 0.5f * bx.w) * ih;
    *(float4*)(sbox + ((size_t)b * K_TOP + (size_t)r) * 4) = make_float4(x1, y1, x2, y2);
    sscore[b * K_TOP + r] = 1.0f / (1.0f + __expf(-logit));
    slabel[b * K_TOP + r] = lab;
  }
}

/* ---------------- K5: greedy NMS, everything LDS-resident ---------------- */
__global__ void __launch_bounds__(1024) k_nms(const float* __restrict__ sbox,
                                              const int* __restrict__ slabel,
                                              unsigned char* __restrict__ keep_g) {
  extern __shared__ char sm[];
  float4* box = (float4*)(sm + L_BOX);
  int* lab = (int*)(sm + L_LAB);
  float* area = (float*)(sm + L_AREA);
  unsigned char* keep = (unsigned char*)(sm + L_KEEP);
  float* red = (float*)(sm + L_RED);
  int b = blockIdx.x, tid = threadIdx.x;
  const float* gb = sbox + (size_t)b * K_TOP * 4;
  const int* gl = slabel + (size_t)b * K_TOP;

  /* CDNA5 async global->LDS staging of the 160 KB box tile + 40 KB labels */
  for (int i = tid; i < K_TOP; i += 1024) {
    async_b128((uint32_t)(uintptr_t)(&box[i]), gb + (size_t)i * 4);
    async_b32((uint32_t)(uintptr_t)(&lab[i]), gl + i);
  }
  wait_async0();
  __syncthreads();

  /* max coordinate -> batched-NMS class offset (max_c + 1) */
  float m = -3.0e38f;
  for (int i = tid; i < K_TOP; i += 1024) {
    float4 v = box[i];
    m = fmaxf(m, fmaxf(fmaxf(v.x, v.y), fmaxf(v.z, v.w)));
  }
  red[tid] = m;
  __syncthreads();
  for (int s = 512; s > 0; s >>= 1) {
    if (tid < s) red[tid] = fmaxf(red[tid], red[tid + s]);
    __syncthreads();
  }
  float off_unit = red[0] + 1.0f;
  __syncthreads();
  for (int i = tid; i < K_TOP; i += 1024) {
    float o = (float)lab[i] * off_unit;
    float4 v = box[i];
    v.x += o; v.y += o; v.z += o; v.w += o;
    box[i] = v;
    area[i] = (v.z - v.x) * (v.w - v.y);
    keep[i] = 1;
  }
  __syncthreads();

  /* greedy scan: outer loop sequential, inner IOU parallel over the block */
  for (int i = 0; i < K_TOP - 1; ++i) {
    if (keep[i]) {
      float4 bi = box[i];
      float ai = area[i];
      for (int j = i + 1 + tid; j < K_TOP; j += 1024) {
        if (keep[j]) {
          float4 bj = box[j];
          float iw = fmaxf(fminf(bi.z, bj.z) - fmaxf(bi.x, bj.x), 0.0f);
          float ihh = fmaxf(fminf(bi.w, bj.w) - fmaxf(bi.y, bj.y), 0.0f);
          float inter = iw * ihh;
          float iou = inter / (ai + area[j] - inter);
          if (iou > IOU_THRESH) keep[j] = 0;
        }
      }
    }
    __syncthreads();
  }
  for (int i = tid; i < K_TOP; i += 1024)
    keep_g[(size_t)b * K_TOP + i] = keep[i];
}

/* ---------------- K6: select first S survivors, write outputs ---------------- */
__global__ void __launch_bounds__(1024) k_out(const float* __restrict__ sbox,
                                              const float* __restrict__ sscore,
                                              const int* __restrict__ slabel,
                                              const unsigned char* __restrict__ keep_g,
                                              float* __restrict__ out, int S) {
  __shared__ uint32_t cnt[1024];
  __shared__ uint32_t pre[1024];
  __shared__ int list[1024];
  __shared__ uint32_t total;
  int b = blockIdx.x, tid = threadIdx.x;
  const unsigned char* kp = keep_g + (size_t)b * K_TOP;
  const int CH = 10;                   /* 1000 threads x 10 = K_TOP */
  int base = tid * CH;
  uint32_t c = 0;
  if (base < K_TOP)
    for (int l = 0; l < CH && base + l < K_TOP; ++l) c += kp[base + l];
  cnt[tid] = c;
  __syncthreads();
  if (tid == 0) {
    uint32_t a = 0;
    for (int t = 0; t < 1024; ++t) { pre[t] = a; a += cnt[t]; }
    total = a;
  }
  __syncthreads();
  if (base < K_TOP) {
    uint32_t r = pre[tid];
    for (int l = 0; l < CH && base + l < K_TOP; ++l) {
      if (kp[base + l]) {
        if (r < (uint32_t)S && r < 1024u) list[r] = base + l;
        ++r;
      }
    }
  }
  __syncthreads();
  uint32_t kept = total;
  float* oboxes = out;
  float* oscores = out + (size_t)BS * S * 4;
  float* olabels = oscores + (size_t)BS * S;
  float* oinds = olabels + (size_t)BS * S;
  for (int k = tid; k < S; k += 1024) {
    float4 bx = make_float4(0.f, 0.f, 0.f, 0.f);
    float sc = 0.f, lb = -1.f, id = -1.f;
    if ((uint32_t)k < kept && k < 1024) {
      int r = list[k];
      bx = *(const float4*)(sbox + ((size_t)b * K_TOP + (size_t)r) * 4);
      sc = sscore[b * K_TOP + r];
      lb = (float)slabel[b * K_TOP + r];
      id = (float)r;
    }
    *(float4*)(oboxes + ((size_t)b * S + (size_t)k) * 4) = bx;
    oscores[b * S + k] = sc;
    olabels[b * S + k] = lb;
    oinds[b * S + k] = id;
  }
}

extern "C" void kernel_launch(void* const* d_in, const int* in_sizes, int n_in,
                              void* d_out, int out_size, void* d_ws, size_t ws_size,
                              hipStream_t stream) {
  const float* logits = (const float*)d_in[0];
  const float* pboxes = (const float*)d_in[1];
  /* d_in[2] = pred_masks: unused by the reference */
  const float* tsz = (const float*)d_in[3];
  float* out = (float*)d_out;
  char* ws = (char*)d_ws;
  int S = out_size / 28; /* 4*(4+1+1+1) floats per selected slot */

  uint32_t* hist = (uint32_t*)(ws + WS_HIST);
  uint32_t* thr = (uint32_t*)(ws + WS_THR);
  uint32_t* selc = (uint32_t*)(ws + WS_SELCNT);
  float* candL = (float*)(ws + WS_CANDL);
  uint32_t* candI = (uint32_t*)(ws + WS_CANDI);
  float* sbox = (float*)(ws + WS_SBOX);
  float* sscore = (float*)(ws + WS_SSCORE);
  int* slabel = (int*)(ws + WS_SLABEL);
  unsigned char* keepg = (unsigned char*)(ws + WS_KEEP);

  /* allow >64KB dynamic LDS for the big-LDS kernels (not a stream op; capture-safe) */
  hipFuncSetAttribute(reinterpret_cast<const void*>(k_sort),
                      hipFuncAttributeMaxDynamicSharedMemorySize, (int)SORT_LDS);
  hipFuncSetAttribute(reinterpret_cast<const void*>(k_nms),
                      hipFuncAttributeMaxDynamicSharedMemorySize, (int)NMS_LDS);

  k_zero<<<256, 256, 0, stream>>>((uint32_t*)ws, (int)(WS_CANDL / 4));
  k_hist<<<dim3(512, BS), 256, 0, stream>>>(logits, hist);
  k_thresh<<<BS, 1024, 0, stream>>>(hist, thr);
  k_compact<<<dim3(512, BS), 256, 0, stream>>>(logits, thr, selc, candL, candI);
  k_sort<<<BS, 1024, SORT_LDS, stream>>>(candL, candI, selc, pboxes, tsz, sbox, sscore, slabel);
  k_nms<<<BS, 1024, NMS_LDS, stream>>>(sbox, slabel, keepg);
  k_out<<<BS, 1024, 0, stream>>>(sbox, sscore, slabel, keepg, out, S);
}